// MoELayer_83880711291366
// MI455X (gfx1250) — compile-verified
//
#include <hip/hip_runtime.h>
#include <hip/hip_bf16.h>
#include <math.h>

// ---------------------------------------------------------------------------
// MoE layer for MI455X (gfx1250): fp32 WMMA (V_WMMA_F32_16X16X4_F32) GEMMs.
// Pipeline: buf0 = x@Wp+bp ; buf1 = buf0@Wv+bv ; buf0 = buf1@Wo+bo (== a)
//           combine = top2-softmax(a@Wg+bg)
//           for e: hid = relu(a@W1[e]+b1[e]); out += combine[:,e]*(hid@W2[e]+b2[e])
//
// GEMM: block tile 128x128, 8 waves (4M x 2N), each wave 32x64 = 8 v8f accums.
// A tile LDS row-major [row][k] stride 36; B tile LDS column-major [col][k]
// stride 36 -> both fragments are contiguous 8-byte ds_load_b64, conflict-free.
// ---------------------------------------------------------------------------

typedef float v2f __attribute__((ext_vector_type(2)));
typedef float v8f __attribute__((ext_vector_type(8)));

#define TM 128          // block tile M
#define TN 128          // block tile N
#define TK 32           // K tile staged in LDS
#define LSTR 36         // LDS stride (floats): 16B-aligned stores, conflict-free

// MODE 0: C = A@B + bias
// MODE 1: C = relu(A@B + bias)
// MODE 2: C += combine[row*10+e] * (A@B + bias)
template <int MODE>
__global__ __launch_bounds__(256) void gemm_wmma_f32(
    const float* __restrict__ A, const float* __restrict__ B,
    const float* __restrict__ bias, float* __restrict__ C,
    int M, int K, int Nn,
    const float* __restrict__ combine, int expertIdx)
{
    __shared__ float As[TM][LSTR];   // [row][k]
    __shared__ float Bs[TN][LSTR];   // [col][k]  (column-major tile)

    const int tid  = threadIdx.x;
    const int m0   = blockIdx.y * TM;
    const int n0   = blockIdx.x * TN;
    const int wave = tid >> 5;
    const int lane = tid & 31;
    const int wm   = (wave & 3) << 5;        // wave's 32-row group: 0,32,64,96
    const int wn   = (wave >> 2) << 6;       // wave's 64-col group: 0,64
    const int lrow = lane & 15;              // M (A) / N (B,C) within 16
    const int lkb  = (lane >> 4) << 1;       // K sub-offset: 0 or 2

    // global->LDS staging indices
    const int arow = tid >> 3;               // 0..31 (+i*32)
    const int acol = (tid & 7) << 2;         // 0,4,...,28
    const int bc   = tid & 127;              // column within tile
    const int bk0  = (tid >> 7) << 4;        // k group: 0 or 16

    v8f acc[2][4];
    #pragma unroll
    for (int i = 0; i < 2; i++)
        #pragma unroll
        for (int j = 0; j < 4; j++) acc[i][j] = {};

    for (int k0 = 0; k0 < K; k0 += TK) {
        // stage A tile [TM x TK], row-major
        #pragma unroll
        for (int i = 0; i < 4; i++) {
            const int r = arow + i * 32;
            const float4 v = *reinterpret_cast<const float4*>(
                &A[(size_t)(m0 + r) * K + (k0 + acol)]);
            *reinterpret_cast<float4*>(&As[r][acol]) = v;
        }
        // stage B tile [TK x TN] transposed into Bs[col][k]
        {
            float tmp[16];
            #pragma unroll
            for (int i = 0; i < 16; i++)
                tmp[i] = B[(size_t)(k0 + bk0 + i) * Nn + (n0 + bc)];
            #pragma unroll
            for (int i = 0; i < 4; i++) {
                float4 v;
                v.x = tmp[4 * i + 0]; v.y = tmp[4 * i + 1];
                v.z = tmp[4 * i + 2]; v.w = tmp[4 * i + 3];
                *reinterpret_cast<float4*>(&Bs[bc][bk0 + 4 * i]) = v;
            }
        }
        __syncthreads();

        #pragma unroll
        for (int kk = 0; kk < TK; kk += 4) {
            // A frags: lane holds A[M=lrow][K=kk+lkb .. +1]   (ISA 16x4 layout)
            v2f a0 = *reinterpret_cast<const v2f*>(&As[wm + lrow][kk + lkb]);
            v2f a1 = *reinterpret_cast<const v2f*>(&As[wm + 16 + lrow][kk + lkb]);
            // B frags: lane holds B[K=kk+lkb .. +1][N=lrow+16j] (contiguous now)
            v2f b[4];
            #pragma unroll
            for (int j = 0; j < 4; j++)
                b[j] = *reinterpret_cast<const v2f*>(
                    &Bs[wn + 16 * j + lrow][kk + lkb]);

            #pragma unroll
            for (int j = 0; j < 4; j++) {
                acc[0][j] = __builtin_amdgcn_wmma_f32_16x16x4_f32(
                    false, a0, false, b[j], (short)0, acc[0][j], false, false);
                acc[1][j] = __builtin_amdgcn_wmma_f32_16x16x4_f32(
                    false, a1, false, b[j], (short)0, acc[1][j], false, false);
            }
        }
        __syncthreads();
    }

    // epilogue: C/D layout -> VGPR jj: row = base + jj + 8*(lane/16), col = lane%16
    const int rb = m0 + wm + ((lane >> 4) << 3);
    const int cb = n0 + wn + lrow;

    #pragma unroll
    for (int i = 0; i < 2; i++) {
        #pragma unroll
        for (int j = 0; j < 4; j++) {
            const int rbase = rb + 16 * i;
            const int col   = cb + 16 * j;
            const float bv  = bias[col];
            #pragma unroll
            for (int jj = 0; jj < 8; jj++) {
                const int r = rbase + jj;
                const size_t idx = (size_t)r * Nn + col;
                const float v = acc[i][j][jj] + bv;
                if (MODE == 0) {
                    C[idx] = v;
                } else if (MODE == 1) {
                    C[idx] = v > 0.f ? v : 0.f;
                } else {
                    const float s = combine[(size_t)r * 10 + expertIdx];
                    C[idx] += s * v;
                }
            }
        }
    }
}

// ---------------------------------------------------------------------------
// Gate: one wave per token. logits = a_row @ Wg + bg; exact top-2 + 2-way
// softmax (identical to full softmax -> top-2 -> renormalize).
// ---------------------------------------------------------------------------
__global__ __launch_bounds__(256) void gate_kernel(
    const float* __restrict__ a, const float* __restrict__ Wg,
    const float* __restrict__ bg, float* __restrict__ combine, int N)
{
    const int wave  = threadIdx.x >> 5;
    const int lane  = threadIdx.x & 31;
    const int token = blockIdx.x * 8 + wave;
    if (token >= N) return;

    const float* __restrict__ row = a + (size_t)token * 1024;
    float acc[10];
    #pragma unroll
    for (int e = 0; e < 10; e++) acc[e] = 0.f;

    for (int d = lane; d < 1024; d += 32) {          // coalesced across the wave
        const float av = row[d];
        const float* __restrict__ wgr = Wg + (size_t)d * 10;
        #pragma unroll
        for (int e = 0; e < 10; e++) acc[e] += av * wgr[e];
    }
    #pragma unroll
    for (int e = 0; e < 10; e++) {
        float v = acc[e];
        #pragma unroll
        for (int off = 16; off > 0; off >>= 1) v += __shfl_xor(v, off, 32);
        acc[e] = v + bg[e];
    }
    if (lane == 0) {
        int i1 = 0; float l1 = acc[0];
        #pragma unroll
        for (int e = 1; e < 10; e++) if (acc[e] > l1) { l1 = acc[e]; i1 = e; }
        int i2 = -1; float l2 = -INFINITY;
        #pragma unroll
        for (int e = 0; e < 10; e++)
            if (e != i1 && acc[e] > l2) { l2 = acc[e]; i2 = e; }
        const float p2 = 1.f / (1.f + expf(l1 - l2));
        const float p1 = 1.f - p2;
        float* crow = combine + (size_t)token * 10;
        #pragma unroll
        for (int e = 0; e < 10; e++) crow[e] = 0.f;
        crow[i1] = p1;
        crow[i2] = p2;
    }
}

__global__ void zero_kernel(float* __restrict__ p, size_t n)
{
    size_t i = (size_t)blockIdx.x * blockDim.x + threadIdx.x;
    const size_t stride = (size_t)gridDim.x * blockDim.x;
    for (; i < n; i += stride) p[i] = 0.f;
}

// ---------------------------------------------------------------------------
extern "C" void kernel_launch(void* const* d_in, const int* in_sizes, int n_in,
                              void* d_out, int out_size, void* d_ws, size_t ws_size,
                              hipStream_t stream)
{
    (void)in_sizes; (void)n_in; (void)out_size; (void)ws_size;

    const float* x  = (const float*)d_in[0];
    // d_in[1] = top_k (always 2 per reference semantics)
    const float* Wp = (const float*)d_in[2];
    const float* bp = (const float*)d_in[3];
    const float* Wv = (const float*)d_in[4];
    const float* bv = (const float*)d_in[5];
    const float* Wo = (const float*)d_in[6];
    const float* bo = (const float*)d_in[7];
    const float* Wg = (const float*)d_in[8];
    const float* bg = (const float*)d_in[9];
    const float* W1 = (const float*)d_in[10];
    const float* b1 = (const float*)d_in[11];
    const float* W2 = (const float*)d_in[12];
    const float* b2 = (const float*)d_in[13];
    float* out = (float*)d_out;

    constexpr int N = 16384, DIN = 1024, D = 1024, HID = 256, E = 10, OUTD = 1024;

    char* ws = (char*)d_ws;
    float* buf0    = (float*)ws;                                  // N*D
    float* buf1    = (float*)(ws + (size_t)N * D * 4);            // N*D
    float* combine = (float*)(ws + 2 * (size_t)N * D * 4);        // N*E
    float* hid     = buf1;  // buf1 is dead after `a` is produced

    const dim3 blk(256);

    // 1) buf0 = x @ Wp + bp
    gemm_wmma_f32<0><<<dim3(D / TN, N / TM), blk, 0, stream>>>(
        x, Wp, bp, buf0, N, DIN, D, nullptr, 0);
    // 2) buf1 = buf0 @ Wv + bv
    gemm_wmma_f32<0><<<dim3(D / TN, N / TM), blk, 0, stream>>>(
        buf0, Wv, bv, buf1, N, D, D, nullptr, 0);
    // 3) buf0 = buf1 @ Wo + bo        (== a)
    gemm_wmma_f32<0><<<dim3(D / TN, N / TM), blk, 0, stream>>>(
        buf1, Wo, bo, buf0, N, D, D, nullptr, 0);

    // 4) gating -> dense combine weights [N, E]
    gate_kernel<<<N / 8, blk, 0, stream>>>(buf0, Wg, bg, combine, N);

    // 5) out = 0 (accumulated across experts)
    zero_kernel<<<2048, blk, 0, stream>>>(out, (size_t)N * OUTD);

    // 6) experts: hid = relu(a@W1[e]+b1[e]);  out += combine[:,e]*(hid@W2[e]+b2[e])
    for (int e = 0; e < E; e++) {
        gemm_wmma_f32<1><<<dim3(HID / TN, N / TM), blk, 0, stream>>>(
            buf0, W1 + (size_t)e * D * HID, b1 + (size_t)e * HID,
            hid, N, D, HID, nullptr, 0);
        gemm_wmma_f32<2><<<dim3(OUTD / TN, N / TM), blk, 0, stream>>>(
            hid, W2 + (size_t)e * HID * OUTD, b2 + (size_t)e * OUTD,
            out, N, HID, OUTD, combine, e);
    }
}